// Cubify_18880676233661
// MI455X (gfx1250) — compile-verified
//
#include <hip/hip_runtime.h>
#include <stdint.h>

// ---------------------------------------------------------------------------
// Cubify for MI455X (gfx1250).
// Pure streaming problem: ~13.5 MB read, ~308 MB written, no matrix math.
// CDNA5 features used: async global->LDS copy (ASYNCcnt) for x-neighbor
// reuse, s_wait_asynccnt, and non-temporal (th:NT) stores for the output
// stream so it does not thrash L2 (input fits L2 and feeds neighbor loads).
// ---------------------------------------------------------------------------

constexpr int N = 4, D = 96, H = 96, W = 96, V = 97;
constexpr long long NVERT      = (long long)V * V * V;            // 912673
constexpr long long VERT_ELEMS = NVERT * 3;                       // 2738019
constexpr long long NVOXFACE   = 6LL * D * H * W;                 // 5308416
constexpr long long FACE_ELEMS = NVOXFACE * 2 * 3;                // 31850496
constexpr long long FACES_OFF  = VERT_ELEMS;                      // 2738019
constexpr long long MASK_OFF   = VERT_ELEMS + FACE_ELEMS;         // 34588515
constexpr long long DIR_STRIDE = (long long)D * H * W * 2;        // 1769472

// Corner-id offsets (oz*97+oy)*97+ox for _QUAD[dir][corner], dir = 0..5.
__constant__ int kCoff[6][4] = {
    {   0,    1,   97,   98},   // dir 0: -z face
    {9409, 9410, 9506, 9507},   // dir 1: +z face
    {9409, 9410,    0,    1},   // dir 2: -y face
    {  97,   98, 9506, 9507},   // dir 3: +y face
    {9409,    0, 9506,   97},   // dir 4: -x face
    {   1, 9410,   98, 9507},   // dir 5: +x face
};

// ---- CDNA5 async global->LDS copy helpers ---------------------------------

typedef __attribute__((address_space(1))) int global_i32;
typedef __attribute__((address_space(3))) int lds_i32;

__device__ __forceinline__ void async_f32_to_lds(const float* gsrc, float* ldst) {
#if __has_builtin(__builtin_amdgcn_global_load_async_to_lds_b32)
    __builtin_amdgcn_global_load_async_to_lds_b32(
        (global_i32*)(uintptr_t)gsrc,
        (lds_i32*)(uint32_t)(uintptr_t)ldst,
        0, 0);
#else
    // Fallback: lds-dest address VGPR, 64-bit global address VGPR pair.
    unsigned l = (unsigned)(uintptr_t)ldst;
    asm volatile("global_load_async_to_lds_b32 %0, %1, off"
                 :: "v"(l), "v"(gsrc)
                 : "memory");
#endif
}

__device__ __forceinline__ void wait_async_zero() {
#if __has_builtin(__builtin_amdgcn_s_wait_asynccnt)
    __builtin_amdgcn_s_wait_asynccnt(0);
#else
    asm volatile("s_wait_asynccnt 0" ::: "memory");
#endif
}

// ---- Kernel 1: verts (batch-invariant, pure index math) -------------------

__global__ void __launch_bounds__(256)
cubify_verts_kernel(float* __restrict__ out) {
    int tid = blockIdx.x * blockDim.x + threadIdx.x;
    if (tid >= (int)NVERT) return;
    int x = tid % V;
    int t = tid / V;
    int y = t % V;
    int z = t / V;
    float* p = out + (long long)tid * 3;
    __builtin_nontemporal_store((float)z - 0.5f, p + 0);
    __builtin_nontemporal_store((float)y - 0.5f, p + 1);
    __builtin_nontemporal_store((float)x - 0.5f, p + 2);
}

// ---- Kernel 2: faces (batch-invariant, pure index math) -------------------
// Layout: (dir, z, y, x, tri, vert); tri0 = corners {0,1,2}, tri1 = {1,2,3}.

__global__ void __launch_bounds__(256)
cubify_faces_kernel(float* __restrict__ out) {
    long long tid = (long long)blockIdx.x * blockDim.x + threadIdx.x;
    if (tid >= NVOXFACE) return;
    int x = (int)(tid % W); long long t = tid / W;
    int y = (int)(t % H);   t /= H;
    int z = (int)(t % D);
    int dir = (int)(t / D);

    int vid = (z * V + y) * V + x;
    float c0 = (float)(vid + kCoff[dir][0]);
    float c1 = (float)(vid + kCoff[dir][1]);
    float c2 = (float)(vid + kCoff[dir][2]);
    float c3 = (float)(vid + kCoff[dir][3]);

    float* p = out + FACES_OFF + tid * 6;   // 2 tris * 3 verts
    __builtin_nontemporal_store(c0, p + 0);
    __builtin_nontemporal_store(c1, p + 1);
    __builtin_nontemporal_store(c2, p + 2);
    __builtin_nontemporal_store(c1, p + 3);
    __builtin_nontemporal_store(c2, p + 4);
    __builtin_nontemporal_store(c3, p + 5);
}

// ---- Kernel 3: face_mask (7-point wraparound stencil) ---------------------
// One block per (n, z, y) row; 96 threads (3 wave32). The 96-float occupancy
// row is staged into LDS with the CDNA5 async copy engine (3x reuse for the
// x-direction neighbors). z/y neighbors are 1:1 -> direct loads (L2 hits).

__global__ void __launch_bounds__(96)
cubify_mask_kernel(const float* __restrict__ vox, float* __restrict__ out) {
    __shared__ float row[W];
    const int x = threadIdx.x;       // 0..95
    const int y = blockIdx.x;        // 0..95
    const int z = blockIdx.y;        // 0..95
    const int n = blockIdx.z;        // 0..3

    const float* rbase = vox + (((long long)(n * D + z)) * H + y) * W;
    async_f32_to_lds(rbase + x, &row[x]);

    // Wraparound neighbors in z and y (jnp.roll semantics).
    const float zm = vox[(((long long)(n * D + (z + D - 1) % D)) * H + y) * W + x];
    const float zp = vox[(((long long)(n * D + (z + 1) % D))     * H + y) * W + x];
    const float ym = vox[(((long long)(n * D + z)) * H + (y + H - 1) % H) * W + x];
    const float yp = vox[(((long long)(n * D + z)) * H + (y + 1) % H)     * W + x];

    wait_async_zero();     // my wave's async copies landed in LDS
    __syncthreads();       // all 3 waves' copies visible

    const float c  = row[x];
    const float xm = row[(x + W - 1) % W];
    const float xp = row[(x + 1) % W];

    const bool oc = c > 0.5f;
    float m[6];
    m[0] = (oc && !(zm > 0.5f)) ? 1.0f : 0.0f;   // nbs = roll(+1, z) -> occ[z-1]
    m[1] = (oc && !(zp > 0.5f)) ? 1.0f : 0.0f;   // roll(-1, z) -> occ[z+1]
    m[2] = (oc && !(ym > 0.5f)) ? 1.0f : 0.0f;
    m[3] = (oc && !(yp > 0.5f)) ? 1.0f : 0.0f;
    m[4] = (oc && !(xm > 0.5f)) ? 1.0f : 0.0f;
    m[5] = (oc && !(xp > 0.5f)) ? 1.0f : 0.0f;

    // face_mask layout: (n, dir, z, y, x, 2). Base element offset is odd, so
    // keep the pair as two b32 NT stores (8B-aligned b64 not guaranteed).
    long long base = MASK_OFF +
        (((((long long)n * 6) * D + z) * H + y) * W + x) * 2;
#pragma unroll
    for (int d = 0; d < 6; ++d) {
        float* p = out + base + (long long)d * DIR_STRIDE;
        __builtin_nontemporal_store(m[d], p);
        __builtin_nontemporal_store(m[d], p + 1);
    }
}

// ---------------------------------------------------------------------------

extern "C" void kernel_launch(void* const* d_in, const int* in_sizes, int n_in,
                              void* d_out, int out_size, void* d_ws, size_t ws_size,
                              hipStream_t stream) {
    const float* vox = (const float*)d_in[0];
    float* out = (float*)d_out;

    (void)in_sizes; (void)n_in; (void)out_size; (void)d_ws; (void)ws_size;

    {   // verts
        const int threads = 256;
        const int blocks = (int)((NVERT + threads - 1) / threads);
        cubify_verts_kernel<<<blocks, threads, 0, stream>>>(out);
    }
    {   // faces
        const int threads = 256;
        const int blocks = (int)((NVOXFACE + threads - 1) / threads);
        cubify_faces_kernel<<<blocks, threads, 0, stream>>>(out);
    }
    {   // face_mask
        dim3 grid(H, D, N);
        dim3 block(W, 1, 1);   // 96 = 3 x wave32
        cubify_mask_kernel<<<grid, block, 0, stream>>>(vox, out);
    }
}